// TGCNAE_50989851738419
// MI455X (gfx1250) — compile-verified
//
#include <hip/hip_runtime.h>
#include <hip/hip_bf16.h>

// ---- problem constants (match reference) ----
#define NN   2000
#define SS   16
#define LL   128
#define FF   3
#define DEG  8
#define NODE_ELEMS (SS * LL)          // 2048 elements per node (H==1)
#define NODE_FLOATS (SS * LL * FF)    // 6144 floats per node input block
#define TILES (NODE_ELEMS / 16)       // 128 M-tiles of 16 per node

typedef __attribute__((ext_vector_type(2))) float v2f;
typedef __attribute__((ext_vector_type(8))) float v8f;

__device__ __forceinline__ float sigm(float x) { return 1.0f / (1.0f + __expf(-x)); }
__device__ __forceinline__ float eluf(float x) { return x > 0.0f ? x : (__expf(x) - 1.0f); }

__global__ __launch_bounds__(128)
void tgcnae_fused_kernel(const float* __restrict__ inputs,
                         const int*   __restrict__ src,
                         const float* __restrict__ w_ih0, const float* __restrict__ w_hh0,
                         const float* __restrict__ b_ih0, const float* __restrict__ b_hh0,
                         const float* __restrict__ w_ih1, const float* __restrict__ w_hh1,
                         const float* __restrict__ b_ih1, const float* __restrict__ b_hh1,
                         const float* __restrict__ We0, const float* __restrict__ be0,
                         const float* __restrict__ We1, const float* __restrict__ be1,
                         const float* __restrict__ Wd0, const float* __restrict__ bd0,
                         const float* __restrict__ Wd1, const float* __restrict__ bd1,
                         float* __restrict__ out)
{
    __shared__ float s_agg[NODE_FLOATS];   // aggregated node features [S][L][F]
    __shared__ float s_y[NODE_ELEMS];      // LSTM(+relu) output       [S][L]
    __shared__ float s_scr[4][16 * 16];    // per-wave transpose scratch

    const int tid  = threadIdx.x;
    const int node = blockIdx.x;
    const int lane = tid & 31;
    const int wv   = tid >> 5;

    // ---------------- Phase A: mean aggregation over DEG in-edges ----------------
    int sbase[DEG];
#pragma unroll
    for (int e = 0; e < DEG; ++e) {
        sbase[e] = src[node * DEG + e] * NODE_FLOATS;   // uniform (scalar) loads
        __builtin_prefetch(inputs + sbase[e] + tid * 4, 0, 0);  // global_prefetch_b8
    }
    const int NV = NODE_FLOATS / 4;   // 1536 float4's
    for (int v = tid; v < NV; v += 128) {
        float4 acc = make_float4(0.f, 0.f, 0.f, 0.f);
#pragma unroll
        for (int e = 0; e < DEG; ++e) {
            const float4 t = *(const float4*)(inputs + sbase[e] + v * 4);
            acc.x += t.x; acc.y += t.y; acc.z += t.z; acc.w += t.w;
        }
        acc.x *= 0.125f; acc.y *= 0.125f; acc.z *= 0.125f; acc.w *= 0.125f;
        ((float4*)s_agg)[v] = acc;
    }
    __syncthreads();

    // ---------------- Phase B: 2-layer LSTM (H==1) scan, one lane per sample ----------------
    if (tid < SS) {
        const int s = tid;
        float wi0[4][3], wh0[4], bb0[4], wi1[4], wh1[4], bb1[4];
#pragma unroll
        for (int g = 0; g < 4; ++g) {
#pragma unroll
            for (int f = 0; f < 3; ++f) wi0[g][f] = w_ih0[g * 3 + f];
            wh0[g] = w_hh0[g];
            bb0[g] = b_ih0[g] + b_hh0[g];
            wi1[g] = w_ih1[g];
            wh1[g] = w_hh1[g];
            bb1[g] = b_ih1[g] + b_hh1[g];
        }
        float h0 = 0.f, c0 = 0.f, h1 = 0.f, c1 = 0.f;
        for (int t = 0; t < LL; ++t) {
            const float* xp = &s_agg[(s * LL + t) * FF];
            const float x0 = xp[0], x1 = xp[1], x2 = xp[2];
            float pre[4];
#pragma unroll
            for (int g = 0; g < 4; ++g)
                pre[g] = wi0[g][0] * x0 + wi0[g][1] * x1 + wi0[g][2] * x2 + bb0[g] + wh0[g] * h0;
            {
                const float ig = sigm(pre[0]), fg = sigm(pre[1]);
                const float gg = tanhf(pre[2]), og = sigm(pre[3]);
                c0 = fg * c0 + ig * gg;
                h0 = og * tanhf(c0);
            }
#pragma unroll
            for (int g = 0; g < 4; ++g)
                pre[g] = wi1[g] * h0 + bb1[g] + wh1[g] * h1;
            {
                const float ig = sigm(pre[0]), fg = sigm(pre[1]);
                const float gg = tanhf(pre[2]), og = sigm(pre[3]);
                c1 = fg * c1 + ig * gg;
                h1 = og * tanhf(c1);
            }
            s_y[s * LL + t] = fmaxf(h1, 0.f);
        }
    }
    __syncthreads();

    // ---------------- Phase C: 1->20->10->20->1 ELU MLP via f32 WMMA ----------------
    // Lane roles per the 16x16x4 f32 VGPR layout:
    //   A: row m = lane&15, K slice kbase = 4c + 2*(lane>>4)
    //   B: col n = lane&15, same K slice
    //   C: VGPR i -> (M = i + 8*(lane>>4), N = lane&15)
    const int hh = lane >> 4;
    const int nn = lane & 15;

    v2f b2[5], we0c[5], be0c[5];
#pragma unroll
    for (int c = 0; c < 5; ++c) {
        const int k0 = 4 * c + 2 * hh;      // <= 19, always valid for K=20
        we0c[c].x = We0[k0];     we0c[c].y = We0[k0 + 1];
        be0c[c].x = be0[k0];     be0c[c].y = be0[k0 + 1];
        b2[c].x = (nn < 10) ? We1[nn * 20 + k0]     : 0.f;   // B2[k][n] = We1[n][k]
        b2[c].y = (nn < 10) ? We1[nn * 20 + k0 + 1] : 0.f;
    }
    v2f b3a[3], b3b[3];
#pragma unroll
    for (int c = 0; c < 3; ++c) {
        const int k0 = 4 * c + 2 * hh;      // valid k < 10
        b3a[c].x = (k0     < 10) ? Wd0[nn * 10 + k0]     : 0.f;         // cols 0..15
        b3a[c].y = (k0 + 1 < 10) ? Wd0[nn * 10 + k0 + 1] : 0.f;
        b3b[c].x = (nn < 4 && k0     < 10) ? Wd0[(16 + nn) * 10 + k0]     : 0.f;  // cols 16..19
        b3b[c].y = (nn < 4 && k0 + 1 < 10) ? Wd0[(16 + nn) * 10 + k0 + 1] : 0.f;
    }
    const float be1n  = (nn < 10) ? be1[nn] : 0.f;
    const float bd0a  = bd0[nn];
    const float bd0b  = (nn < 4) ? bd0[16 + nn] : 0.f;
    const float wd1a  = Wd1[nn];
    const float wd1b  = (nn < 4) ? Wd1[16 + nn] : 0.f;
    const float bd1v  = bd1[0];

    float* scr = &s_scr[wv][0];

    for (int tile = wv; tile < TILES; tile += 4) {
        const float yv = s_y[tile * 16 + nn];    // A row m = nn

        // layer1 (1->20) built directly in A layout, layer2 (20->10) via 5 WMMA
        v8f acc = {};
#pragma unroll
        for (int c = 0; c < 5; ++c) {
            v2f a;
            a.x = eluf(yv * we0c[c].x + be0c[c].x);
            a.y = eluf(yv * we0c[c].y + be0c[c].y);
            acc = __builtin_amdgcn_wmma_f32_16x16x4_f32(false, a, false, b2[c],
                                                        (short)0, acc, false, false);
        }

        // z2 = elu(acc + be1): spill C layout -> A layout through per-wave LDS scratch
#pragma unroll
        for (int i = 0; i < 8; ++i) {
            const float z = eluf(acc[i] + be1n);   // cols n>=10 stay exactly 0
            const int row = i + 8 * hh;
            scr[row * 16 + nn] = z;
        }
        asm volatile("s_wait_dscnt 0x0" ::: "memory");

        // layer3 (10->20) via 3 WMMA per N-tile
        v8f a3a = {}, a3b = {};
#pragma unroll
        for (int c = 0; c < 3; ++c) {
            const int k0 = 4 * c + 2 * hh;
            v2f a;
            a.x = scr[nn * 16 + k0];
            a.y = scr[nn * 16 + k0 + 1];
            a3a = __builtin_amdgcn_wmma_f32_16x16x4_f32(false, a, false, b3a[c],
                                                        (short)0, a3a, false, false);
            a3b = __builtin_amdgcn_wmma_f32_16x16x4_f32(false, a, false, b3b[c],
                                                        (short)0, a3b, false, false);
        }

        // layer4 (20->1): per-lane partial dot, then 16-lane xor butterfly over N
        float part[8];
#pragma unroll
        for (int i = 0; i < 8; ++i)
            part[i] = eluf(a3a[i] + bd0a) * wd1a + eluf(a3b[i] + bd0b) * wd1b;
#pragma unroll
        for (int mask = 1; mask <= 8; mask <<= 1) {
#pragma unroll
            for (int i = 0; i < 8; ++i)
                part[i] += __shfl_xor(part[i], mask, 32);
        }
        if (nn == 0) {
            const int base = node * NODE_ELEMS + tile * 16 + 8 * hh;
            float4 o0, o1;
            o0.x = eluf(part[0] + bd1v); o0.y = eluf(part[1] + bd1v);
            o0.z = eluf(part[2] + bd1v); o0.w = eluf(part[3] + bd1v);
            o1.x = eluf(part[4] + bd1v); o1.y = eluf(part[5] + bd1v);
            o1.z = eluf(part[6] + bd1v); o1.w = eluf(part[7] + bd1v);
            *(float4*)(out + base)     = o0;
            *(float4*)(out + base + 4) = o1;
        }
        asm volatile("s_wait_dscnt 0x0" ::: "memory");  // WAR guard before next tile's spill
    }
}

extern "C" void kernel_launch(void* const* d_in, const int* in_sizes, int n_in,
                              void* d_out, int out_size, void* d_ws, size_t ws_size,
                              hipStream_t stream) {
    (void)in_sizes; (void)n_in; (void)out_size; (void)d_ws; (void)ws_size;
    const float* inputs = (const float*)d_in[0];
    const int*   src    = (const int*)d_in[1];
    // d_in[2] = dst (structurally arange(E)//DEG, unused)
    const float* w_ih0 = (const float*)d_in[3];
    const float* w_hh0 = (const float*)d_in[4];
    const float* b_ih0 = (const float*)d_in[5];
    const float* b_hh0 = (const float*)d_in[6];
    const float* w_ih1 = (const float*)d_in[7];
    const float* w_hh1 = (const float*)d_in[8];
    const float* b_ih1 = (const float*)d_in[9];
    const float* b_hh1 = (const float*)d_in[10];
    const float* We0 = (const float*)d_in[11];
    const float* be0 = (const float*)d_in[12];
    const float* We1 = (const float*)d_in[13];
    const float* be1 = (const float*)d_in[14];
    const float* Wd0 = (const float*)d_in[15];
    const float* bd0 = (const float*)d_in[16];
    const float* Wd1 = (const float*)d_in[17];
    const float* bd1 = (const float*)d_in[18];
    float* out = (float*)d_out;

    tgcnae_fused_kernel<<<dim3(NN), dim3(128), 0, stream>>>(
        inputs, src,
        w_ih0, w_hh0, b_ih0, b_hh0,
        w_ih1, w_hh1, b_ih1, b_hh1,
        We0, be0, We1, be1, Wd0, bd0, Wd1, bd1,
        out);
}